// TorchCompileBatchTopK_76192719831802
// MI455X (gfx1250) — compile-verified
//
#include <hip/hip_runtime.h>
#include <stdint.h>

// BatchTopK on MI455X (gfx1250). Memory-bound radix-select:
//  K1: 4096-bin histogram of |ranking| bit patterns (async global->LDS stream)
//  K2: find coarse threshold bin b1 + remaining rank r1
//  K3: compact candidates (bucket == b1) to workspace
//  K4: refine 19 low bits over candidates -> exact threshold T + tie budget
//  K5: apply threshold, write output (x/out non-temporal; ranking temporal so
//      its re-reads hit the 192MB L2)
//
// Assumes B = 1024 rows (k_total = k_per_token * 1024), per the reference.

#define HIST_BINS 4096
#define K1_THREADS 256

typedef int   v4i __attribute__((vector_size(16)));
typedef float v4f __attribute__((vector_size(16)));

// ---------------- gfx1250 async-to-LDS helpers (guarded) ----------------
#if defined(__HIP_DEVICE_COMPILE__) && __has_builtin(__builtin_amdgcn_global_load_async_to_lds_b128)
#define HAVE_ASYNC_LDS 1
#else
#define HAVE_ASYNC_LDS 0
#endif

__device__ __forceinline__ void async_copy16(const void* g, void* l) {
#if HAVE_ASYNC_LDS
  const v4i* gv = (const v4i*)g;
  v4i* lv = (v4i*)l;
  __builtin_amdgcn_global_load_async_to_lds_b128(
      (__attribute__((address_space(1))) v4i*)(v4i*)gv,
      (__attribute__((address_space(3))) v4i*)lv,
      0, 0);
#else
  *(float4*)l = *(const float4*)g;
#endif
}

__device__ __forceinline__ void wait_async_le1() {
#if HAVE_ASYNC_LDS
#if __has_builtin(__builtin_amdgcn_s_wait_asynccnt)
  __builtin_amdgcn_s_wait_asynccnt(1);
#else
  asm volatile("s_wait_asynccnt 1" ::: "memory");
#endif
#endif
}

__device__ __forceinline__ void wait_async_le0() {
#if HAVE_ASYNC_LDS
#if __has_builtin(__builtin_amdgcn_s_wait_asynccnt)
  __builtin_amdgcn_s_wait_asynccnt(0);
#else
  asm volatile("s_wait_asynccnt 0" ::: "memory");
#endif
#endif
}

__device__ __forceinline__ unsigned abs_bits(float f) {
  return __float_as_uint(f) & 0x7fffffffu;
}

// ---------------- K1: global histogram over bits[30:19] ----------------
__global__ void __launch_bounds__(K1_THREADS)
k_hist(const float* __restrict__ rank, unsigned* __restrict__ hist, long long n4) {
  __shared__ unsigned lh[HIST_BINS];
  __shared__ __align__(16) float stage[2][K1_THREADS * 4];

  for (int i = threadIdx.x; i < HIST_BINS; i += K1_THREADS) lh[i] = 0;
  __syncthreads();

  const float4* r4 = (const float4*)rank;
  const long long stride = (long long)gridDim.x * K1_THREADS;
  long long idx = (long long)blockIdx.x * K1_THREADS + threadIdx.x;
  float* myslot0 = &stage[0][threadIdx.x * 4];
  float* myslot1 = &stage[1][threadIdx.x * 4];
  int buf = 0;

  if (idx < n4) async_copy16(r4 + idx, myslot0);

  while (idx < n4) {
    long long nxt = idx + stride;
    if (nxt < n4) {
      async_copy16(r4 + nxt, buf ? myslot0 : myslot1);
      wait_async_le1();               // current stage done, next in flight
    } else {
      wait_async_le0();
    }
    const float4 v = *(const float4*)(buf ? myslot1 : myslot0);
    atomicAdd(&lh[abs_bits(v.x) >> 19], 1u);
    atomicAdd(&lh[abs_bits(v.y) >> 19], 1u);
    atomicAdd(&lh[abs_bits(v.z) >> 19], 1u);
    atomicAdd(&lh[abs_bits(v.w) >> 19], 1u);
    buf ^= 1;
    idx = nxt;
  }

  __syncthreads();
  for (int i = threadIdx.x; i < HIST_BINS; i += K1_THREADS) {
    unsigned c = lh[i];
    if (c) atomicAdd(&hist[i], c);
  }
}

// ---------------- K2: coarse threshold bin ----------------
// sc[0]=b1  sc[1]=count_above  sc[2]=r1  sc[3]=cand_count  sc[4]=T
// sc[5]=tie_budget  sc[6]=tie_counter
__global__ void __launch_bounds__(256)
k_select1(const unsigned* __restrict__ hist, unsigned* __restrict__ sc,
          const int* __restrict__ kptr) {
  __shared__ unsigned csum[256];
  __shared__ unsigned before[256];
  const unsigned t = threadIdx.x;
  const unsigned kt = (unsigned)(*kptr) * 1024u;  // k_total = k_per_token * B

  // chunk t covers 16 bins descending from the top
  const unsigned hi = (HIST_BINS - 1) - t * 16;
  unsigned loc[16];
  unsigned s = 0;
  #pragma unroll
  for (int j = 0; j < 16; ++j) { loc[j] = hist[hi - j]; s += loc[j]; }
  csum[t] = s;
  __syncthreads();
  if (t == 0) {
    unsigned acc = 0;
    for (int i = 0; i < 256; ++i) { before[i] = acc; acc += csum[i]; }
  }
  __syncthreads();
  const unsigned b = before[t];
  if (b < kt && b + s >= kt) {
    unsigned acc = b;
    #pragma unroll
    for (int j = 0; j < 16; ++j) {
      if (acc + loc[j] >= kt) {
        sc[0] = hi - j;      // b1
        sc[1] = acc;         // strictly-above count
        sc[2] = kt - acc;    // remaining rank within b1 (>=1)
        break;
      }
      acc += loc[j];
    }
  }
}

// ---------------- K3: compact candidates in bin b1 ----------------
__global__ void __launch_bounds__(256)
k_compact(const float* __restrict__ rank, unsigned* __restrict__ sc,
          unsigned* __restrict__ cand, unsigned cap, long long n4) {
  const unsigned b1 = sc[0];
  const float4* r4 = (const float4*)rank;
  const long long stride = (long long)gridDim.x * blockDim.x;
  for (long long i = (long long)blockIdx.x * blockDim.x + threadIdx.x; i < n4; i += stride) {
    const float4 v = r4[i];   // temporal: should mostly hit L2 after K1
    const unsigned a0 = abs_bits(v.x), a1 = abs_bits(v.y);
    const unsigned a2 = abs_bits(v.z), a3 = abs_bits(v.w);
    if ((a0 >> 19) == b1) { unsigned p = atomicAdd(&sc[3], 1u); if (p < cap) cand[p] = a0; }
    if ((a1 >> 19) == b1) { unsigned p = atomicAdd(&sc[3], 1u); if (p < cap) cand[p] = a1; }
    if ((a2 >> 19) == b1) { unsigned p = atomicAdd(&sc[3], 1u); if (p < cap) cand[p] = a2; }
    if ((a3 >> 19) == b1) { unsigned p = atomicAdd(&sc[3], 1u); if (p < cap) cand[p] = a3; }
  }
}

// ---------------- K4: refine low 19 bits over candidates ----------------
__global__ void __launch_bounds__(1024)
k_select2(unsigned* __restrict__ sc, const unsigned* __restrict__ cand, unsigned cap) {
  __shared__ unsigned h[1024];
  __shared__ unsigned sh_sel, sh_rem;
  const unsigned tid = threadIdx.x;
  const unsigned n = min(sc[3], cap);
  unsigned r = sc[2];
  const unsigned b1 = sc[0];

  // round A: bits [18:9] (1024 bins)
  h[tid] = 0;
  __syncthreads();
  for (unsigned i = tid; i < n; i += 1024) atomicAdd(&h[(cand[i] >> 9) & 1023u], 1u);
  __syncthreads();
  if (tid == 0) {
    unsigned acc = 0; int s = 1023;
    for (; s > 0; --s) { if (acc + h[s] >= r) break; acc += h[s]; }
    sh_sel = (unsigned)s;
    sh_rem = (r > acc) ? (r - acc) : 1u;
  }
  __syncthreads();
  const unsigned s1 = sh_sel;
  r = sh_rem;
  __syncthreads();

  // round B: bits [8:0] (512 bins), prefix-filtered
  h[tid] = 0;
  __syncthreads();
  for (unsigned i = tid; i < n; i += 1024) {
    const unsigned v = cand[i];
    if (((v >> 9) & 1023u) == s1) atomicAdd(&h[v & 511u], 1u);
  }
  __syncthreads();
  if (tid == 0) {
    unsigned acc = 0; int s = 511;
    for (; s > 0; --s) { if (acc + h[s] >= r) break; acc += h[s]; }
    const unsigned s2 = (unsigned)s;
    const unsigned rem = (r > acc) ? (r - acc) : 1u;
    sc[4] = (b1 << 19) | (s1 << 9) | s2;  // exact 31-bit threshold T
    sc[5] = rem;                          // tie budget at T
  }
}

// ---------------- K5: apply threshold ----------------
__global__ void __launch_bounds__(256)
k_apply(const float* __restrict__ x, const float* __restrict__ rank,
        float* __restrict__ out, unsigned* __restrict__ sc, long long n4) {
  const unsigned T = sc[4];
  const v4f* x4 = (const v4f*)x;
  const v4f* r4 = (const v4f*)rank;
  v4f* o4 = (v4f*)out;
  const long long stride = (long long)gridDim.x * blockDim.x;
  for (long long i = (long long)blockIdx.x * blockDim.x + threadIdx.x; i < n4; i += stride) {
    // x is consumed once: non-temporal load so it doesn't evict ranking from L2
    const v4f xv = __builtin_nontemporal_load(x4 + i);
    // ranking re-read: temporal, mostly L2-resident after K1/K3
    const v4f rv = r4[i];
    v4f ov;
    #pragma unroll
    for (int j = 0; j < 4; ++j) {
      const unsigned a = abs_bits(rv[j]);
      float o = 0.0f;
      if (a > T) {
        o = xv[j];
      } else if (a == T) {
        if (atomicAdd(&sc[6], 1u) < sc[5]) o = xv[j];
      }
      ov[j] = o;
    }
    // output is write-once: non-temporal store, don't pollute L2
    __builtin_nontemporal_store(ov, o4 + i);
  }
}

// ---------------- host launcher ----------------
extern "C" void kernel_launch(void* const* d_in, const int* in_sizes, int n_in,
                              void* d_out, int out_size, void* d_ws, size_t ws_size,
                              hipStream_t stream) {
  const float* x    = (const float*)d_in[0];
  const float* rank = (const float*)d_in[1];
  const int*   kptr = (const int*)d_in[2];
  float* out = (float*)d_out;

  const long long N  = (long long)in_sizes[0];  // B*F = 67108864
  const long long n4 = N / 4;

  unsigned* ws   = (unsigned*)d_ws;
  unsigned* hist = ws;                 // [0, 4096)
  unsigned* sc   = ws + HIST_BINS;     // scalars [4096, 4103)
  unsigned* cand = ws + 4352;          // candidate buffer
  size_t avail = (ws_size / 4 > 4352) ? (ws_size / 4 - 4352) : 0;
  unsigned cap = (unsigned)((avail < (size_t)(1u << 22)) ? avail : (size_t)(1u << 22));

  // zero histogram + scalar scratch every call (graph-capture safe)
  (void)hipMemsetAsync(ws, 0, 4352 * sizeof(unsigned), stream);

  k_hist   <<<4096, K1_THREADS, 0, stream>>>(rank, hist, n4);
  k_select1<<<1,    256,        0, stream>>>(hist, sc, kptr);
  k_compact<<<4096, 256,        0, stream>>>(rank, sc, cand, cap, n4);
  k_select2<<<1,    1024,       0, stream>>>(sc, cand, cap);
  k_apply  <<<8192, 256,        0, stream>>>(x, rank, out, sc, n4);
}